// GCN_78176994721834
// MI455X (gfx1250) — compile-verified
//
#include <hip/hip_runtime.h>

typedef __attribute__((ext_vector_type(2))) float v2f;
typedef __attribute__((ext_vector_type(8))) float v8f;

#define IN_DIM 128
#define HIDDEN 128
#define NCLS   64
#define BN_EPS 1e-5f

// ---------------- degree / norm ----------------
__global__ void k_init_deg(unsigned* __restrict__ deg, int n) {
    int i = blockIdx.x * blockDim.x + threadIdx.x;
    if (i < n) deg[i] = 1u;                 // self loop contributes 1
}

__global__ void k_count_deg(const int* __restrict__ dst, unsigned* __restrict__ deg, int ne) {
    int i = blockIdx.x * blockDim.x + threadIdx.x;
    if (i < ne) atomicAdd(&deg[dst[i]], 1u);
}

__global__ void k_dinv(const unsigned* __restrict__ deg, float* __restrict__ dinv, int n) {
    int i = blockIdx.x * blockDim.x + threadIdx.x;
    if (i < n) {
        float d = (float)deg[i];
        dinv[i] = (d > 0.f) ? rsqrtf(d) : 0.f;
    }
}

// ---------------- WMMA GEMM:  H[nrows x NCOLS] = X[nrows x 128] * W[128 x NCOLS] ----------
// grid.x = ceil(nrows/32); blockDim.x = (NCOLS/16)*32   (one wave per 16-col tile)
// Each wave computes TWO 16x16 output tiles (rows m0..m0+15 and m0+16..m0+31) so every
// B-operand load feeds two independent V_WMMA_F32_16X16X4_F32 chains.
template <int NCOLS>
__global__ __launch_bounds__(256) void k_gemm_wmma(const float* __restrict__ X,
                                                   const float* __restrict__ W,
                                                   float* __restrict__ H,
                                                   int nrows) {
    const int wave = threadIdx.x >> 5;
    const int lane = threadIdx.x & 31;
    const int hl   = lane >> 4;          // wave half: K pairs {0,1} vs {2,3}
    const int l    = lane & 15;
    const int m0   = blockIdx.x << 5;    // 32 rows per block
    const int n0   = wave << 4;

    // A rows for the two M-tiles (clamped so loads stay in range; stores guarded below)
    int r0 = m0 + l;       if (r0 >= nrows) r0 = nrows - 1;
    int r1 = m0 + 16 + l;  if (r1 >= nrows) r1 = nrows - 1;
    const float* __restrict__ x0 = X + (size_t)r0 * IN_DIM + (hl << 1);
    const float* __restrict__ x1 = X + (size_t)r1 * IN_DIM + (hl << 1);
    const float* __restrict__ wp = W + (size_t)(hl << 1) * NCOLS + n0 + l;

    v8f acc0 = {};
    v8f acc1 = {};
#pragma unroll 8
    for (int k = 0; k < IN_DIM; k += 4) {
        v2f a0, a1, b;
        a0.x = x0[k];
        a0.y = x0[k + 1];
        a1.x = x1[k];
        a1.y = x1[k + 1];
        b.x  = wp[k * NCOLS];            // immediate offsets: NCOLS is compile-time
        b.y  = wp[(k + 1) * NCOLS];
        acc0 = __builtin_amdgcn_wmma_f32_16x16x4_f32(false, a0, false, b,
                                                     (short)0, acc0, false, false);
        acc1 = __builtin_amdgcn_wmma_f32_16x16x4_f32(false, a1, false, b,
                                                     (short)0, acc1, false, false);
    }

    // C/D layout: VGPR r -> row base + hl*8 + r, col n0 + l
    float* __restrict__ h0 = H + (size_t)(m0 + (hl << 3)) * NCOLS + n0 + l;
    float* __restrict__ h1 = h0 + (size_t)16 * NCOLS;
    if (m0 + 32 <= nrows) {              // uniform fast path: no per-lane guards
#pragma unroll
        for (int r = 0; r < 8; ++r) {
            h0[r * NCOLS] = acc0[r];
            h1[r * NCOLS] = acc1[r];
        }
    } else {                             // remainder block only
#pragma unroll
        for (int r = 0; r < 8; ++r) {
            if (m0 + (hl << 3) + r < nrows)      h0[r * NCOLS] = acc0[r];
            if (m0 + 16 + (hl << 3) + r < nrows) h1[r * NCOLS] = acc1[r];
        }
    }
}

// ---------------- self-loop init passes (no atomics) ----------------
__global__ void k_self1(const float* __restrict__ H, const float* __restrict__ dinv,
                        float* __restrict__ out, int n) {
    long i = blockIdx.x * (long)blockDim.x + threadIdx.x;
    if (i >= (long)n * HIDDEN) return;
    int node = (int)(i >> 7);
    float dv = dinv[node];
    out[i] = H[i] * dv * dv;
}

__global__ void k_self2(const float* __restrict__ H, const float* __restrict__ dinv,
                        const float* __restrict__ b2, float* __restrict__ out, int n) {
    long i = blockIdx.x * (long)blockDim.x + threadIdx.x;
    if (i >= (long)n * NCLS) return;
    int node = (int)(i >> 6);
    int f    = (int)(i & (NCLS - 1));
    float dv = dinv[node];
    out[i] = H[i] * dv * dv + b2[f];
}

// ---------------- edge scatter: one wave per edge ----------------
template <int F>
__global__ void k_edge_scatter(const float* __restrict__ H, const int* __restrict__ src,
                               const int* __restrict__ dst, const float* __restrict__ dinv,
                               float* __restrict__ out, int ne) {
    long gid = blockIdx.x * (long)blockDim.x + threadIdx.x;
    int e = (int)(gid >> 5);
    if (e >= ne) return;
    int lane = (int)(gid & 31);
    int s = src[e], d = dst[e];
    float nrm = dinv[s] * dinv[d];
    const float* __restrict__ hs = H + (size_t)s * F;
    float* __restrict__ od = out + (size_t)d * F;

    if constexpr (F == 128) {
        float4 v = reinterpret_cast<const float4*>(hs)[lane];   // coalesced b128 gather
        int f0 = lane << 2;
        atomicAdd(&od[f0 + 0], v.x * nrm);
        atomicAdd(&od[f0 + 1], v.y * nrm);
        atomicAdd(&od[f0 + 2], v.z * nrm);
        atomicAdd(&od[f0 + 3], v.w * nrm);
    } else {
        float2 v = reinterpret_cast<const float2*>(hs)[lane];   // coalesced b64 gather
        int f0 = lane << 1;
        atomicAdd(&od[f0 + 0], v.x * nrm);
        atomicAdd(&od[f0 + 1], v.y * nrm);
    }
}

// ---------------- bias + BatchNorm(eval) + ReLU, in place ----------------
__global__ void k_bnrelu(float* __restrict__ a, const float* __restrict__ b1,
                         const float* __restrict__ gamma, const float* __restrict__ beta,
                         const float* __restrict__ mean, const float* __restrict__ var, int n) {
    long i = blockIdx.x * (long)blockDim.x + threadIdx.x;
    if (i >= (long)n * HIDDEN) return;
    int f = (int)(i & (HIDDEN - 1));
    float t = (a[i] + b1[f] - mean[f]) * rsqrtf(var[f] + BN_EPS) * gamma[f] + beta[f];
    a[i] = fmaxf(t, 0.f);
}

extern "C" void kernel_launch(void* const* d_in, const int* in_sizes, int n_in,
                              void* d_out, int out_size, void* d_ws, size_t ws_size,
                              hipStream_t stream) {
    const float* x     = (const float*)d_in[0];
    const int*   ei    = (const int*)  d_in[1];
    const float* W1    = (const float*)d_in[2];
    const float* b1    = (const float*)d_in[3];
    const float* gamma = (const float*)d_in[4];
    const float* beta  = (const float*)d_in[5];
    const float* rmean = (const float*)d_in[6];
    const float* rvar  = (const float*)d_in[7];
    const float* W2    = (const float*)d_in[8];
    const float* b2    = (const float*)d_in[9];

    const int N = in_sizes[0] / IN_DIM;
    const int E = in_sizes[1] / 2;
    const int* src = ei;
    const int* dst = ei + E;

    // workspace carve-up
    float*    H    = (float*)d_ws;                      // N*128 (H1, later H2)
    float*    AGG  = H + (size_t)N * HIDDEN;            // N*128 (agg1 -> bn/relu in place)
    unsigned* deg  = (unsigned*)(AGG + (size_t)N * HIDDEN);
    float*    dinv = (float*)(deg + N);
    float*    out  = (float*)d_out;

    const int T = 256;
    // graph normalization
    k_init_deg <<<(N + T - 1) / T, T, 0, stream>>>(deg, N);
    k_count_deg<<<(E + T - 1) / T, T, 0, stream>>>(dst, deg, E);
    k_dinv     <<<(N + T - 1) / T, T, 0, stream>>>(deg, dinv, N);

    // layer 1: H1 = X @ W1
    k_gemm_wmma<HIDDEN><<<(N + 31) / 32, (HIDDEN / 16) * 32, 0, stream>>>(x, W1, H, N);

    const long tot1  = (long)N * HIDDEN;
    const long etoks = (long)E * 32;
    k_self1<<<(int)((tot1 + T - 1) / T), T, 0, stream>>>(H, dinv, AGG, N);
    k_edge_scatter<128><<<(int)((etoks + T - 1) / T), T, 0, stream>>>(H, src, dst, dinv, AGG, E);
    k_bnrelu<<<(int)((tot1 + T - 1) / T), T, 0, stream>>>(AGG, b1, gamma, beta, rmean, rvar, N);

    // layer 2: H2 = relu(...) @ W2  (reuse H buffer)
    k_gemm_wmma<NCLS><<<(N + 31) / 32, (NCLS / 16) * 32, 0, stream>>>(AGG, W2, H, N);

    const long tot2 = (long)N * NCLS;
    k_self2<<<(int)((tot2 + T - 1) / T), T, 0, stream>>>(H, dinv, b2, out, N);
    k_edge_scatter<64><<<(int)((etoks + T - 1) / T), T, 0, stream>>>(H, src, dst, dinv, out, E);
}